// LearnedGateDeltaModel_3204045603590
// MI455X (gfx1250) — compile-verified
//
#include <hip/hip_runtime.h>
#include <hip/hip_bf16.h>

#define H      64
#define H2     128
#define VOCAB  64
#define BATCH  128
#define SEQLEN 4096
#define GH     16

typedef __attribute__((ext_vector_type(2))) float v2f;
typedef __attribute__((ext_vector_type(8))) float v8f;

static __device__ __forceinline__ v8f wmma_f32_16x16x4(v2f a, v2f b, v8f c) {
    // 8 args: (neg_a, A, neg_b, B, c_mod, C, reuse_a, reuse_b)
    return __builtin_amdgcn_wmma_f32_16x16x4_f32(false, a, false, b, (short)0, c, false, false);
}

// ---------------------------------------------------------------------------
// Phase 1: per-token featurization  hn = LN(x + relu(x@W1+b1)@W2+b2)
// One wave (32 threads) per 16-token tile. Weights read from global (L2/WGP$
// resident: W1+W2+embed = 80KB). f32 WMMA 16x16x4 tiles.
// ---------------------------------------------------------------------------
__global__ __launch_bounds__(32) void tok_ff_ln_kernel(
    const int* __restrict__ seq, const float* __restrict__ embed,
    const float* __restrict__ w1, const float* __restrict__ b1,
    const float* __restrict__ w2, const float* __restrict__ b2,
    const float* __restrict__ lng, const float* __restrict__ lnb,
    float* __restrict__ hn)
{
    __shared__ float sY1[16 * H2];   // relu(x@W1+b1) tile, row-major 16x128
    __shared__ int   sIdx[16];       // vocab indices for the 16 tokens

    const int lane = threadIdx.x;
    const int l16  = lane & 15;
    const int hs   = lane >> 4;                  // selects K pair within 4-step
    const size_t n0 = (size_t)blockIdx.x * 16;   // first token of tile

    if (lane < 16) sIdx[lane] = seq[n0 + lane];
    __syncthreads();

    const int    srow = sIdx[l16];
    const float* xrow = embed + (size_t)srow * H;

    // ---- GEMM1: (16xH) @ (H x H2), K stepped by 4 -------------------------
    v8f acc1[8];
    #pragma unroll
    for (int nt = 0; nt < 8; ++nt)
        #pragma unroll
        for (int r = 0; r < 8; ++r) acc1[nt][r] = 0.0f;

    #pragma unroll
    for (int kk = 0; kk < H / 4; ++kk) {
        const int kb = 4 * kk + 2 * hs;
        v2f a; a[0] = xrow[kb]; a[1] = xrow[kb + 1];
        #pragma unroll
        for (int nt = 0; nt < 8; ++nt) {
            const int col = nt * 16 + l16;
            v2f bf;
            bf[0] = w1[(size_t)kb * H2 + col];
            bf[1] = w1[(size_t)(kb + 1) * H2 + col];
            acc1[nt] = wmma_f32_16x16x4(a, bf, acc1[nt]);
        }
    }

    // bias + relu, re-major through LDS (C layout -> row major)
    #pragma unroll
    for (int nt = 0; nt < 8; ++nt)
        #pragma unroll
        for (int r = 0; r < 8; ++r) {
            const int row = r + 8 * hs;
            const int col = nt * 16 + l16;
            const float v = acc1[nt][r] + b1[col];
            sY1[row * H2 + col] = v > 0.0f ? v : 0.0f;
        }
    __syncthreads();

    // ---- GEMM2: (16xH2) @ (H2 x H) ---------------------------------------
    v8f acc2[4];
    #pragma unroll
    for (int nt = 0; nt < 4; ++nt)
        #pragma unroll
        for (int r = 0; r < 8; ++r) acc2[nt][r] = 0.0f;

    #pragma unroll
    for (int kk = 0; kk < H2 / 4; ++kk) {
        const int kb = 4 * kk + 2 * hs;
        v2f a; a[0] = sY1[l16 * H2 + kb]; a[1] = sY1[l16 * H2 + kb + 1];
        #pragma unroll
        for (int nt = 0; nt < 4; ++nt) {
            const int col = nt * 16 + l16;
            v2f bf;
            bf[0] = w2[(size_t)kb * H + col];
            bf[1] = w2[(size_t)(kb + 1) * H + col];
            acc2[nt] = wmma_f32_16x16x4(a, bf, acc2[nt]);
        }
    }

    // ---- residual + LayerNorm + store ------------------------------------
    #pragma unroll
    for (int r = 0; r < 8; ++r) {
        const int row = r + 8 * hs;
        const int er  = sIdx[row];
        const float* xr = embed + (size_t)er * H;
        float vals[4];
        float s1 = 0.0f, s2 = 0.0f;
        #pragma unroll
        for (int nt = 0; nt < 4; ++nt) {
            const int col = nt * 16 + l16;
            const float v = acc2[nt][r] + b2[col] + xr[col];
            vals[nt] = v; s1 += v; s2 += v * v;
        }
        #pragma unroll
        for (int m = 1; m < 16; m <<= 1) {
            s1 += __shfl_xor(s1, m, 32);
            s2 += __shfl_xor(s2, m, 32);
        }
        const float mu   = s1 * (1.0f / (float)H);
        const float var  = s2 * (1.0f / (float)H) - mu * mu;
        const float rstd = rsqrtf(var + 1e-5f);
        float* op = hn + (n0 + row) * H;
        #pragma unroll
        for (int nt = 0; nt < 4; ++nt) {
            const int col = nt * 16 + l16;
            op[col] = (vals[nt] - mu) * rstd * lng[col] + lnb[col];
        }
    }
}

// ---------------------------------------------------------------------------
// Phase 2: sequential delta-rule scan. One workgroup (96 thr = 3 waves) per
// batch. Threads 0..63 each own row i of M in 64 VGPRs; threads 64..79 (wave
// 2, lanes 0..15) run the gate MLP in parallel. k double-buffered in LDS with
// a 2-step-deep register prefetch so the global load latency is hidden.
// Two barriers per step.
// ---------------------------------------------------------------------------
__global__ __launch_bounds__(96) void scan_kernel(
    const float* __restrict__ hn,
    const float* __restrict__ gw1, const float* __restrict__ gb1,
    const float* __restrict__ gw2, const float* __restrict__ gb2,
    float* __restrict__ ctxws, float* __restrict__ wrs_out)
{
    __shared__ float sK[2][H];
    __shared__ float sGW1[H * GH];
    __shared__ float sGate;

    const int tid = threadIdx.x;
    const int b   = blockIdx.x;
    const float* hb = hn + (size_t)b * SEQLEN * H;

    for (int i = tid; i < H * GH; i += 96) sGW1[i] = gw1[i];

    const bool isRow  = (tid < H);
    const bool isGate = (tid >= H) && (tid < H + GH);
    const int  gl     = tid - H;               // gate lane 0..15 (wave-2 lanes)

    const float gb1v = isGate ? gb1[gl] : 0.0f;
    const float gw2v = isGate ? gw2[gl] : 0.0f;
    const float gb2v = gb2[0];

    float Mrow[H];
    #pragma unroll
    for (int j = 0; j < H; ++j) Mrow[j] = 0.0f;
    float cnt  = 0.0f;
    float kreg = 0.0f;

    if (isRow) {
        sK[0][tid] = hb[tid];          // k_0
        kreg       = hb[H + tid];      // prefetch k_1
    }
    __syncthreads();

    for (int t = 0; t < SEQLEN - 1; ++t) {
        const int cur = t & 1;
        const int nxt = cur ^ 1;

        float vp = 0.0f, k2 = 0.0f;
        if (isRow) {
            // publish prefetched k_{t+1}; issue load for k_{t+2}
            sK[nxt][tid] = kreg;
            const int tp = (t + 2 < SEQLEN) ? (t + 2) : (SEQLEN - 1);
            kreg = hb[(size_t)tp * H + tid];

            // v_pred[i] = M[i,:] . k ; |k|^2 (LDS broadcast, float4 reads)
            #pragma unroll
            for (int j4 = 0; j4 < H / 4; ++j4) {
                const float4 kq = *reinterpret_cast<const float4*>(&sK[cur][4 * j4]);
                vp = fmaf(Mrow[4 * j4 + 0], kq.x, vp); k2 = fmaf(kq.x, kq.x, k2);
                vp = fmaf(Mrow[4 * j4 + 1], kq.y, vp); k2 = fmaf(kq.y, kq.y, k2);
                vp = fmaf(Mrow[4 * j4 + 2], kq.z, vp); k2 = fmaf(kq.z, kq.z, k2);
                vp = fmaf(Mrow[4 * j4 + 3], kq.w, vp); k2 = fmaf(kq.w, kq.w, k2);
            }
        } else if (isGate) {
            // gate MLP runs on its own wave, parallel to the row work
            float z = gb1v;
            #pragma unroll
            for (int i = 0; i < H; ++i) z = fmaf(sK[cur][i], sGW1[i * GH + gl], z);
            z = fmaxf(z, 0.0f);
            float zw = z * gw2v;
            #pragma unroll
            for (int m = 1; m < GH; m <<= 1) zw += __shfl_xor(zw, m, 32);
            if (gl == 0) {
                const float g = 1.0f / (1.0f + expf(-(zw + gb2v)));
                sGate = g;
                cnt += (g > 0.5f) ? 1.0f : 0.0f;
            }
        }
        __syncthreads();   // sGate + sK[nxt] published

        if (isRow) {
            const float g     = sGate;
            const float denom = k2 + 1e-6f;
            const float dv    = sK[cur][tid] - vp / denom;
            const float coef  = g * dv;
            #pragma unroll
            for (int j4 = 0; j4 < H / 4; ++j4) {
                const float4 kq = *reinterpret_cast<const float4*>(&sK[cur][4 * j4]);
                Mrow[4 * j4 + 0] = fmaf(coef, kq.x, Mrow[4 * j4 + 0]);
                Mrow[4 * j4 + 1] = fmaf(coef, kq.y, Mrow[4 * j4 + 1]);
                Mrow[4 * j4 + 2] = fmaf(coef, kq.z, Mrow[4 * j4 + 2]);
                Mrow[4 * j4 + 3] = fmaf(coef, kq.w, Mrow[4 * j4 + 3]);
            }
        }
        __syncthreads();   // close all reads of sK[cur] / sGate for this step
    }

    // q = hn[b, L-1] ended up in sK[(SEQLEN-1)&1] == sK[1]
    if (isRow) {
        float ctx = 0.0f;
        #pragma unroll
        for (int j = 0; j < H; ++j) ctx = fmaf(Mrow[j], sK[1][j], ctx);
        ctxws[b * H + tid] = ctx;
    }
    if (tid == H)
        atomicAdd(wrs_out, cnt * (1.0f / ((float)BATCH * (float)(SEQLEN - 1))));
}

// ---------------------------------------------------------------------------
// Phase 3: logits = (ctx@read_w + read_b)@out_w + out_b. One wave per 16-row
// batch tile; same f32 WMMA tiling as phase 1.
// ---------------------------------------------------------------------------
__global__ __launch_bounds__(32) void head_kernel(
    const float* __restrict__ ctxws,
    const float* __restrict__ rw, const float* __restrict__ rb,
    const float* __restrict__ ow, const float* __restrict__ ob,
    float* __restrict__ logits)
{
    __shared__ float sT[16 * H];

    const int lane = threadIdx.x;
    const int l16  = lane & 15;
    const int hs   = lane >> 4;
    const int b0   = blockIdx.x * 16;

    v8f acc[4];
    #pragma unroll
    for (int nt = 0; nt < 4; ++nt)
        #pragma unroll
        for (int r = 0; r < 8; ++r) acc[nt][r] = 0.0f;

    const float* crow = ctxws + (size_t)(b0 + l16) * H;
    #pragma unroll
    for (int kk = 0; kk < H / 4; ++kk) {
        const int kb = 4 * kk + 2 * hs;
        v2f a; a[0] = crow[kb]; a[1] = crow[kb + 1];
        #pragma unroll
        for (int nt = 0; nt < 4; ++nt) {
            const int col = nt * 16 + l16;
            v2f bf;
            bf[0] = rw[(size_t)kb * H + col];
            bf[1] = rw[(size_t)(kb + 1) * H + col];
            acc[nt] = wmma_f32_16x16x4(a, bf, acc[nt]);
        }
    }
    #pragma unroll
    for (int nt = 0; nt < 4; ++nt)
        #pragma unroll
        for (int r = 0; r < 8; ++r) {
            const int row = r + 8 * hs;
            const int col = nt * 16 + l16;
            sT[row * H + col] = acc[nt][r] + rb[col];
        }
    __syncthreads();

    v8f acc2[4];
    #pragma unroll
    for (int nt = 0; nt < 4; ++nt)
        #pragma unroll
        for (int r = 0; r < 8; ++r) acc2[nt][r] = 0.0f;

    #pragma unroll
    for (int kk = 0; kk < H / 4; ++kk) {
        const int kb = 4 * kk + 2 * hs;
        v2f a; a[0] = sT[l16 * H + kb]; a[1] = sT[l16 * H + kb + 1];
        #pragma unroll
        for (int nt = 0; nt < 4; ++nt) {
            const int col = nt * 16 + l16;
            v2f bf;
            bf[0] = ow[(size_t)kb * H + col];
            bf[1] = ow[(size_t)(kb + 1) * H + col];
            acc2[nt] = wmma_f32_16x16x4(a, bf, acc2[nt]);
        }
    }
    #pragma unroll
    for (int nt = 0; nt < 4; ++nt)
        #pragma unroll
        for (int r = 0; r < 8; ++r) {
            const int row = r + 8 * hs;
            const int col = nt * 16 + l16;
            logits[(size_t)(b0 + row) * VOCAB + col] = acc2[nt][r] + ob[col];
        }
}

__global__ void init_out_kernel(float* __restrict__ out) {
    if (threadIdx.x == 0 && blockIdx.x == 0) out[BATCH * VOCAB] = 0.0f;
}

extern "C" void kernel_launch(void* const* d_in, const int* in_sizes, int n_in,
                              void* d_out, int out_size, void* d_ws, size_t ws_size,
                              hipStream_t stream) {
    const int*   seq    = (const int*)d_in[0];
    const float* embed  = (const float*)d_in[1];
    const float* ff_w1  = (const float*)d_in[2];
    const float* ff_b1  = (const float*)d_in[3];
    const float* ff_w2  = (const float*)d_in[4];
    const float* ff_b2  = (const float*)d_in[5];
    const float* ln_g   = (const float*)d_in[6];
    const float* ln_b   = (const float*)d_in[7];
    const float* gw1    = (const float*)d_in[8];
    const float* gb1    = (const float*)d_in[9];
    const float* gw2    = (const float*)d_in[10];
    const float* gb2    = (const float*)d_in[11];
    const float* read_w = (const float*)d_in[12];
    const float* read_b = (const float*)d_in[13];
    const float* out_w  = (const float*)d_in[14];
    const float* out_b  = (const float*)d_in[15];

    float* out = (float*)d_out;
    float* hn  = (float*)d_ws;                       // B*L*H f32 = 128 MB
    float* ctx = hn + (size_t)BATCH * SEQLEN * H;    // B*H f32

    init_out_kernel<<<1, 32, 0, stream>>>(out);
    tok_ff_ln_kernel<<<(BATCH * SEQLEN) / 16, 32, 0, stream>>>(
        seq, embed, ff_w1, ff_b1, ff_w2, ff_b2, ln_g, ln_b, hn);
    scan_kernel<<<BATCH, 96, 0, stream>>>(hn, gw1, gb1, gw2, gb2, ctx,
                                          out + BATCH * VOCAB);
    head_kernel<<<BATCH / 16, 32, 0, stream>>>(ctx, read_w, read_b, out_w,
                                               out_b, out);
}